// PatchedVLLMKVCache_5781025980799
// MI455X (gfx1250) — compile-verified
//
#include <hip/hip_runtime.h>
#include <stdint.h>

// ---------------------------------------------------------------------------
// PatchedVLLMKVCache for MI455X (gfx1250)
//   out[slot_row] = token_owned(slot) ? clip(input[tok]/s_in, +-240)*s_out
//                                     : cache[slot]*s_out
// Memory-bound streaming kernel (1.0 GiB traffic -> ~43 us @ 23.3 TB/s).
// Fused single pass over the 512 MiB output using an inverse slot->token map
// in d_ws; 4 KiB slot-rows staged through LDS with the CDNA5 async DMA engine
// (global_load_async_to_lds_b128 / s_wait_asynccnt), 4-buffer ring, prefetch
// depth 3 (1.5 KB of reads in flight per wave to help fill the HBM pipe).
// ---------------------------------------------------------------------------

typedef float v4f __attribute__((ext_vector_type(4)));

static constexpr int   kBlockSize   = 128;                     // tokens per cache block
static constexpr int   kRowFloats   = 8 * 128;                 // heads * head_dim = 1024
static constexpr int   kRowV4       = kRowFloats / 4;          // 256 float4 per slot row
static constexpr int   kNumSlots    = 1024 * 128;              // 131072 slot rows
static constexpr int   kSlotsPerWG  = 8;                       // rows pipelined per block
static constexpr float kFp8Max      = 240.0f;

// ---- CDNA5 async global->LDS helpers (ASYNCcnt tracked) -------------------
__device__ __forceinline__ void async_load_b128(uint32_t lds_byte_addr, uint64_t gaddr) {
    // per-lane: copies 16 bytes from global address gaddr to LDS address lds_byte_addr, non-blocking
    asm volatile("global_load_async_to_lds_b128 %0, %1, off"
                 :: "v"(lds_byte_addr), "v"(gaddr) : "memory");
}
__device__ __forceinline__ void wait_async_le_3() {
    asm volatile("s_wait_asynccnt 0x3" ::: "memory");
}
__device__ __forceinline__ void wait_async_le_0() {
    asm volatile("s_wait_asynccnt 0x0" ::: "memory");
}
__device__ __forceinline__ void wait_ds_0() {
    asm volatile("s_wait_dscnt 0x0" ::: "memory");
}

__device__ __forceinline__ uint32_t lds_addr_of(const void* p) {
    // generic LDS pointer: shared aperture in the high half, LDS offset in the low; low 32 bits
    // are exactly what the async-LDS VDST operand wants.
    return (uint32_t)(uintptr_t)p;
}

// ---- map construction ------------------------------------------------------
__global__ void __launch_bounds__(256)
init_map_kernel(int* __restrict__ map) {
    int i = blockIdx.x * 256 + threadIdx.x;
    if (i < kNumSlots) map[i] = -1;
}

__global__ void __launch_bounds__(256)
build_map_kernel(const int* __restrict__ bi, const int* __restrict__ bo,
                 int* __restrict__ map, int ntok) {
    int t = blockIdx.x * 256 + threadIdx.x;
    if (t < ntok) map[bi[t] * kBlockSize + bo[t]] = t;
}

// ---- fused main pass -------------------------------------------------------
// 256 threads = 8 wave32s; wave w owns bytes [w*512, w*512+512) of every row,
// lane l the 16 B at w*512 + l*16  ==> float4 index within row == threadIdx.x.
// Each wave only ever touches its own LDS segment -> per-wave ASYNCcnt/DScnt
// waits are sufficient, no barriers anywhere.
__global__ void __launch_bounds__(256)
fused_scatter_dequant_kernel(const float* __restrict__ input,
                             const float* __restrict__ cache,
                             const int*   __restrict__ map,
                             const float* __restrict__ p_si,
                             const float* __restrict__ p_so,
                             float*       __restrict__ out) {
    __shared__ v4f buf[4][256];                   // 4 x 4 KiB row ring
    const int tid = threadIdx.x;                  // float4 index within a row

    const float si     = p_si[0];
    const float so     = p_so[0];
    const float inv_si = 1.0f / si;

    const v4f* __restrict__ inV  = (const v4f*)input;
    const v4f* __restrict__ ccV  = (const v4f*)cache;
    v4f*       __restrict__ outV = (v4f*)out;

    const int slot0 = blockIdx.x * kSlotsPerWG;

    bool tok[4];

    // prologue: stage rows 0..2 (prefetch depth 3)
#pragma unroll
    for (int r = 0; r < 3; ++r) {
        const int s = slot0 + r;
        const int t = map[s];                     // uniform -> scalar load
        tok[r] = (t >= 0);
        const v4f* src = tok[r] ? (inV + (size_t)t * kRowV4)
                                : (ccV + (size_t)s * kRowV4);
        async_load_b128(lds_addr_of(&buf[r][tid]),
                        (uint64_t)(uintptr_t)(src + tid));
    }

#pragma unroll
    for (int i = 0; i < kSlotsPerWG; ++i) {
        if (i + 3 < kSlotsPerWG) {
            const int s  = slot0 + i + 3;
            const int t  = map[s];
            const bool tk = (t >= 0);
            tok[(i + 3) & 3] = tk;
            const v4f* src = tk ? (inV + (size_t)t * kRowV4)
                                : (ccV + (size_t)s * kRowV4);
            wait_ds_0();   // this wave's read of buf[(i+3)&3] (iter i-1) retired
            async_load_b128(lds_addr_of(&buf[(i + 3) & 3][tid]),
                            (uint64_t)(uintptr_t)(src + tid));
            wait_async_le_3();                    // <=3 outstanding: row i landed
        } else {
            wait_async_le_0();                    // drain tail (strictly safe)
        }

        v4f v = buf[i & 3][tid];                  // ds_load_b128
        v4f r;
        if (tok[i & 3]) {
            v4f q = v * inv_si;
            q.x = fminf(fmaxf(q.x, -kFp8Max), kFp8Max);
            q.y = fminf(fmaxf(q.y, -kFp8Max), kFp8Max);
            q.z = fminf(fmaxf(q.z, -kFp8Max), kFp8Max);
            q.w = fminf(fmaxf(q.w, -kFp8Max), kFp8Max);
            r = q * so;
        } else {
            r = v * so;
        }
        __builtin_nontemporal_store(r, outV + (size_t)(slot0 + i) * kRowV4 + tid);
    }
}

// ---- fallback (d_ws too small): classic 2-pass -----------------------------
__global__ void __launch_bounds__(256)
scale_cache_kernel(const float* __restrict__ cache,
                   const float* __restrict__ p_so,
                   float* __restrict__ out) {
    const float so = p_so[0];
    size_t i = (size_t)blockIdx.x * 256 + threadIdx.x;    // one float4 each
    const v4f* ccV = (const v4f*)cache;
    v4f v = __builtin_nontemporal_load(ccV + i);
    __builtin_nontemporal_store(v * so, (v4f*)out + i);
}

__global__ void __launch_bounds__(256)
scatter_tokens_kernel(const float* __restrict__ input,
                      const int* __restrict__ bi, const int* __restrict__ bo,
                      const float* __restrict__ p_si, const float* __restrict__ p_so,
                      float* __restrict__ out) {
    const int tok  = blockIdx.x;
    const int tid  = threadIdx.x;
    const float si = p_si[0], so = p_so[0];
    const float inv_si = 1.0f / si;
    const size_t slot = (size_t)bi[tok] * kBlockSize + bo[tok];

    v4f v = ((const v4f*)input)[(size_t)tok * kRowV4 + tid];
    v4f q = v * inv_si;
    q.x = fminf(fmaxf(q.x, -kFp8Max), kFp8Max);
    q.y = fminf(fmaxf(q.y, -kFp8Max), kFp8Max);
    q.z = fminf(fmaxf(q.z, -kFp8Max), kFp8Max);
    q.w = fminf(fmaxf(q.w, -kFp8Max), kFp8Max);
    __builtin_nontemporal_store(q * so, (v4f*)out + slot * kRowV4 + tid);
}

// ---------------------------------------------------------------------------
extern "C" void kernel_launch(void* const* d_in, const int* in_sizes, int n_in,
                              void* d_out, int out_size, void* d_ws, size_t ws_size,
                              hipStream_t stream) {
    const float* input = (const float*)d_in[0];
    const float* cache = (const float*)d_in[1];
    const int*   bi    = (const int*)d_in[2];
    const int*   bo    = (const int*)d_in[3];
    const float* p_si  = (const float*)d_in[4];
    const float* p_so  = (const float*)d_in[5];
    float*       out   = (float*)d_out;
    const int    ntok  = in_sizes[2];

    if (ws_size >= (size_t)kNumSlots * sizeof(int)) {
        int* map = (int*)d_ws;
        init_map_kernel<<<kNumSlots / 256, 256, 0, stream>>>(map);
        build_map_kernel<<<(ntok + 255) / 256, 256, 0, stream>>>(bi, bo, map, ntok);
        fused_scatter_dequant_kernel<<<kNumSlots / kSlotsPerWG, 256, 0, stream>>>(
            input, cache, map, p_si, p_so, out);
    } else {
        // 2-pass fallback: scale everything, then overwrite token rows.
        scale_cache_kernel<<<(kNumSlots * kRowV4) / 256, 256, 0, stream>>>(cache, p_so, out);
        scatter_tokens_kernel<<<ntok, 256, 0, stream>>>(input, bi, bo, p_si, p_so, out);
    }
}